// RoIHeadTemplate_62002147885308
// MI455X (gfx1250) — compile-verified
//
#include <hip/hip_runtime.h>
#include <hip/hip_bf16.h>
#include <stdint.h>

// Problem constants (match reference)
#define B_      16
#define N_      32768
#define C_      3
#define PRE_    4096
#define POST_   512
#define IOU_T   0.7f
#define THREADS 1024
#define CH_     2048          // boxes per TDM-staged chunk in Phase A
#define NCHUNK  (N_ / CH_)    // 16

// ---------------- LDS layout (dynamic shared memory) ----------------
// [0, 256K)  : 32768 x u64 sort keys           (Phase A/sort)
//              aliased after sort by candidate arrays (Phase B..D):
//              x1[4096] y1[4096] x2[4096] y2[4096] area[4096] score[4096] idx[4096]  (112K)
// [256K ...) : suppression bitmap (128 u32), sel[512], comm[16],
//              double-buffered TDM stage (2 * CH_*3 f32 = 48K)
#define SZ_KEYS   ((size_t)N_ * 8)                 // 262144
#define OFF_X1    0
#define OFF_Y1    (PRE_ * 4)
#define OFF_X2    (2 * PRE_ * 4)
#define OFF_Y2    (3 * PRE_ * 4)
#define OFF_AREA  (4 * PRE_ * 4)
#define OFF_SC    (5 * PRE_ * 4)
#define OFF_IDX   (6 * PRE_ * 4)
#define OFF_SUPP  SZ_KEYS                          // 262144
#define OFF_SEL   (OFF_SUPP + 128 * 4)             // 262656
#define OFF_COMM  (OFF_SEL + POST_ * 4)            // 264704
#define OFF_STAGE (OFF_COMM + 64)                  // 264768
#define STAGE_BYTES ((size_t)CH_ * C_ * 4)         // 24576 per buffer
#define LDS_TOTAL (OFF_STAGE + 2 * STAGE_BYTES)    // 313920 (< 320KB WGP LDS)

#if defined(__gfx1250__) && __has_builtin(__builtin_amdgcn_tensor_load_to_lds)
#define USE_TDM 1
#else
#define USE_TDM 0
#endif

#if USE_TDM
typedef unsigned int v4u __attribute__((ext_vector_type(4)));
typedef int          v4i __attribute__((ext_vector_type(4)));
typedef int          v8i __attribute__((ext_vector_type(8)));

// Issue (without waiting) a 1-D TDM copy: n_dwords f32 elements from global
// memory into LDS at lds_byte_addr. D# built per CDNA5 ISA §8 (group0:
// count/lds_addr/global_addr/type=2; group1: data_size=4B,
// tensor_dim0=tile_dim0=n, dim1=1, stride0=n). TENSORcnt tracks completion.
__device__ __forceinline__ void tdm_issue_1d(const void* gsrc, unsigned lds_byte_addr,
                                             unsigned n_dwords) {
  unsigned long long ga = (unsigned long long)(uintptr_t)gsrc;
  v4u g0;
  g0.x = 1u;                                              // count=1 (user descriptor)
  g0.y = lds_byte_addr;                                   // lds_addr (bytes)
  g0.z = (unsigned)(ga & 0xFFFFFFFFu);                    // global_addr[31:0]
  g0.w = (unsigned)((ga >> 32) & 0x01FFFFFFu) | (2u << 30); // global_addr[56:32] | type=2
  v8i g1;
  g1[0] = (int)(2u << 16);                                // wg_mask=0, data_size=4B
  g1[1] = (int)((n_dwords & 0xFFFFu) << 16);              // tensor_dim0[15:0]
  g1[2] = (int)(((n_dwords >> 16) & 0xFFFFu) | (1u << 16)); // tensor_dim0[31:16] | tensor_dim1=1
  g1[3] = (int)((n_dwords & 0xFFFFu) << 16);              // tensor_dim1 hi=0 | tile_dim0=n
  g1[4] = 1;                                              // tile_dim1=1, tile_dim2=0
  g1[5] = (int)n_dwords;                                  // tensor_dim0_stride[31:0]
  g1[6] = 0;                                              // stride0 hi | stride1 lo
  g1[7] = 0;                                              // stride1 hi
  v4i g2 = {0, 0, 0, 0};
  v4i g3 = {0, 0, 0, 0};
#if __clang_major__ >= 23
  v8i gx = {0, 0, 0, 0, 0, 0, 0, 0};
  __builtin_amdgcn_tensor_load_to_lds(g0, g1, g2, g3, gx, 0);
#else
  __builtin_amdgcn_tensor_load_to_lds(g0, g1, g2, g3, 0);
#endif
}

__device__ __forceinline__ void tdm_wait_le1() {
#if __has_builtin(__builtin_amdgcn_s_wait_tensorcnt)
  __builtin_amdgcn_s_wait_tensorcnt(1);
#else
  asm volatile("s_wait_tensorcnt 0x1" ::: "memory");
#endif
}
__device__ __forceinline__ void tdm_wait_le0() {
#if __has_builtin(__builtin_amdgcn_s_wait_tensorcnt)
  __builtin_amdgcn_s_wait_tensorcnt(0);
#else
  asm volatile("s_wait_tensorcnt 0x0" ::: "memory");
#endif
}
#endif  // USE_TDM

__global__ __launch_bounds__(THREADS) void
nms_roihead_kernel(const float* __restrict__ boxes,   // (B,N,7)
                   const float* __restrict__ cls,     // (B,N,3)
                   float* __restrict__ out) {
  extern __shared__ char smem[];
  unsigned long long* keys = (unsigned long long*)smem;
  float*    x1   = (float*)(smem + OFF_X1);
  float*    y1   = (float*)(smem + OFF_Y1);
  float*    x2   = (float*)(smem + OFF_X2);
  float*    y2   = (float*)(smem + OFF_Y2);
  float*    area = (float*)(smem + OFF_AREA);
  float*    csc  = (float*)(smem + OFF_SC);
  int*      cidx = (int*)  (smem + OFF_IDX);
  unsigned* supp = (unsigned*)(smem + OFF_SUPP);
  int*      sel  = (int*)  (smem + OFF_SEL);
  int*      comm = (int*)  (smem + OFF_COMM);
  float*    stage= (float*)(smem + OFF_STAGE);   // two buffers of CH_*3 floats
  (void)stage;

  const int tid = threadIdx.x;
  const int b   = blockIdx.x;

  // ---------------- Phase A: scores = max over classes; build sort keys ---
  // Double-buffered TDM pipeline: DMA of chunk c+1 overlaps key-pack of c.
#if USE_TDM
  if (tid < 32) {   // wave 0 drives the Tensor Data Mover (EXEC-independent DMA)
    tdm_issue_1d(cls + (size_t)b * N_ * C_,
                 (unsigned)(uintptr_t)(smem + OFF_STAGE), (unsigned)(CH_ * C_));
  }
#endif
  for (int chunk = 0; chunk < NCHUNK; ++chunk) {
    const float* src = cls + ((size_t)b * N_ + (size_t)chunk * CH_) * C_;
#if USE_TDM
    const float* buf = (const float*)(smem + OFF_STAGE + (size_t)(chunk & 1) * STAGE_BYTES);
    if (tid < 32) {
      if (chunk + 1 < NCHUNK) {
        tdm_issue_1d(cls + ((size_t)b * N_ + (size_t)(chunk + 1) * CH_) * C_,
                     (unsigned)(uintptr_t)(smem + OFF_STAGE +
                                           (size_t)((chunk + 1) & 1) * STAGE_BYTES),
                     (unsigned)(CH_ * C_));
        tdm_wait_le1();    // chunk's DMA complete (in-order); next stays in flight
      } else {
        tdm_wait_le0();
      }
    }
    __syncthreads();  // TDM data visible in LDS to all waves
#endif
    for (int i = tid; i < CH_; i += THREADS) {
#if USE_TDM
      float c0 = buf[i * 3 + 0], c1 = buf[i * 3 + 1], c2 = buf[i * 3 + 2];
#else
      __builtin_prefetch(src + i * 3, 0, 1);            // global_prefetch_b8
      float c0 = src[i * 3 + 0], c1 = src[i * 3 + 1], c2 = src[i * 3 + 2];
#endif
      float sc = fmaxf(c0, fmaxf(c1, c2));
      unsigned bits = __float_as_uint(sc);              // scores in (0,1): bits order = value order
      int gi = chunk * CH_ + i;
      // ascending u64 sort => descending score, lower index first on ties (top_k semantics)
      keys[gi] = ((unsigned long long)(~bits) << 32) | (unsigned)gi;
    }
#if USE_TDM
    __syncthreads();  // buffer (chunk&1) reused two iterations later; this fences readers
#endif
    (void)src;
  }

  // ---------------- In-LDS bitonic full sort of 32768 keys (256 KB) -------
  for (unsigned k = 2; k <= (unsigned)N_; k <<= 1) {
    for (unsigned j = k >> 1; j > 0; j >>= 1) {
      __syncthreads();
      for (unsigned t = tid; t < (unsigned)N_; t += THREADS) {
        unsigned p = t ^ j;
        if (p > t) {
          unsigned long long a = keys[t], c = keys[p];
          bool up = ((t & k) == 0);
          bool sw = up ? (a > c) : (a < c);
          if (sw) { keys[t] = c; keys[p] = a; }
        }
      }
    }
  }
  __syncthreads();

  // ---------------- Phase B: decode top-4096, build BEV footprints --------
  {
    unsigned long long kk[PRE_ / THREADS];
    unsigned idx[PRE_ / THREADS];
    float sc[PRE_ / THREADS], bx[PRE_ / THREADS], by[PRE_ / THREADS],
          bdx[PRE_ / THREADS], bdy[PRE_ / THREADS];
#pragma unroll
    for (int m = 0; m < PRE_ / THREADS; ++m) {
      kk[m]  = keys[tid + m * THREADS];
      idx[m] = (unsigned)kk[m];
      sc[m]  = __uint_as_float(~(unsigned)(kk[m] >> 32));
      const float* bp = boxes + ((size_t)b * N_ + idx[m]) * 7;
      __builtin_prefetch(bp, 0, 3);                     // global_prefetch_b8
      bx[m] = bp[0]; by[m] = bp[1]; bdx[m] = bp[3]; bdy[m] = bp[4];
    }
    __syncthreads();  // keys region is about to be overwritten by footprint arrays
#pragma unroll
    for (int m = 0; m < PRE_ / THREADS; ++m) {
      int c = tid + m * THREADS;
      float hx = bdx[m] * 0.5f, hy = bdy[m] * 0.5f;
      float a1 = bx[m] - hx, a2 = bx[m] + hx, b1 = by[m] - hy, b2 = by[m] + hy;
      x1[c] = a1; x2[c] = a2; y1[c] = b1; y2[c] = b2;
      area[c] = (a2 - a1) * (b2 - b1);
      csc[c] = sc[m]; cidx[c] = (int)idx[m];
    }
  }
  if (tid < 128) supp[tid] = 0u;
  if (tid == 0) { comm[0] = 0; comm[1] = 0; comm[2] = -1; }
  __syncthreads();

  // ---------------- Phase C: greedy NMS via suppression bitmap ------------
  for (;;) {
    if (tid == 0) {
      int cur = comm[0], ns = comm[1], pick = -1;
      if (ns < POST_) {
        while (cur < PRE_ && ((supp[cur >> 5] >> (cur & 31)) & 1u)) ++cur;
        if (cur < PRE_) { pick = cur; sel[ns] = cur; comm[1] = ns + 1; cur = cur + 1; }
      }
      comm[0] = cur; comm[2] = pick;
    }
    __syncthreads();
    int i = comm[2];
    if (i < 0) break;
    float xi1 = x1[i], yi1 = y1[i], xi2 = x2[i], yi2 = y2[i], ai = area[i];
    for (int j = i + 1 + tid; j < PRE_; j += THREADS) {
      float iw = fminf(xi2, x2[j]) - fmaxf(xi1, x1[j]);
      float ih = fminf(yi2, y2[j]) - fmaxf(yi1, y1[j]);
      iw = fmaxf(iw, 0.f); ih = fmaxf(ih, 0.f);
      float inter = iw * ih;
      float iou = inter / (ai + area[j] - inter + 1e-8f);
      if (iou >= IOU_T) atomicOr(&supp[j >> 5], 1u << (j & 31));  // ds_or_b32
    }
    __syncthreads();
  }

  // ---------------- Phase D: gather outputs (rois | scores | labels) ------
  int nsel = comm[1];
  float* rois = out;
  float* rsc  = out + (size_t)B_ * POST_ * 7;
  float* rlb  = rsc + (size_t)B_ * POST_;
  for (int k = tid; k < POST_; k += THREADS) {
    float o[7] = {0.f, 0.f, 0.f, 0.f, 0.f, 0.f, 0.f};
    float sc = 0.f, lb = 1.f;   // pad: rois=0, score=0, label=0+1
    if (k < nsel) {
      int c = sel[k];
      int oi = cidx[c];
      const float* bp = boxes + ((size_t)b * N_ + oi) * 7;
#pragma unroll
      for (int m = 0; m < 7; ++m) o[m] = bp[m];
      sc = csc[c];
      const float* cp = cls + ((size_t)b * N_ + oi) * C_;
      float c0 = cp[0], c1 = cp[1], c2 = cp[2];
      int l = 0; float best = c0;
      if (c1 > best) { best = c1; l = 1; }
      if (c2 > best) { best = c2; l = 2; }
      lb = (float)(l + 1);
    }
    float* ro = rois + ((size_t)b * POST_ + k) * 7;
#pragma unroll
    for (int m = 0; m < 7; ++m) ro[m] = o[m];
    rsc[(size_t)b * POST_ + k] = sc;
    rlb[(size_t)b * POST_ + k] = lb;
  }
}

extern "C" void kernel_launch(void* const* d_in, const int* in_sizes, int n_in,
                              void* d_out, int out_size, void* d_ws, size_t ws_size,
                              hipStream_t stream) {
  (void)in_sizes; (void)n_in; (void)d_ws; (void)ws_size; (void)out_size;
  const float* boxes = (const float*)d_in[0];   // (B,N,7) f32
  const float* cls   = (const float*)d_in[1];   // (B,N,3) f32
  float* out = (float*)d_out;                   // rois|scores|labels, 73728 f32
  nms_roihead_kernel<<<dim3(B_), dim3(THREADS), LDS_TOTAL, stream>>>(boxes, cls, out);
}